// HyCoRec_10093173145840
// MI455X (gfx1250) — compile-verified
//
#include <hip/hip_runtime.h>
#include <math.h>

// ---------------------------------------------------------------------------
// Problem constants (match reference)
// ---------------------------------------------------------------------------
#define DD    128
#define HH    8
#define NN    20000
#define NNZ_  500000
#define EE    5000
#define CC    50
#define NENT_ 50000

typedef float v2f __attribute__((ext_vector_type(2)));
typedef float v8f __attribute__((ext_vector_type(8)));

// ---------------------------------------------------------------------------
// WMMA GEMM:  Y[M,128] = X[M,128] @ W[128,128] (+ bias[128])
//
// Block = 256 threads = 8 waves, covers 128 output rows.
//   - W (64 KB) is staged in LDS once per block (vectorized b128 copy),
//     shared by all 8 waves -> 8x less L2 weight traffic than per-tile reads.
//   - Wave w owns the 16-row stripe [blockIdx*128 + 16w, ...+16).
//     Its A fragments (64 floats/lane) are loaded once into VGPRs and reused
//     across all 8 column tiles (256 V_WMMA_F32_16X16X4_F32 per wave).
// Per-lane layouts (ISA 7.12.2):
//   A 16x4 : lane L -> M = L%16, K = vgpr + 2*(L/16)
//   B 4x16 : lane L -> N = L%16, K = vgpr + 2*(L/16)
//   C 16x16: lane L -> N = L%16, M = vgpr + 8*(L/16)
// EXEC stays all-ones at every WMMA: reads clamped, stores masked.
// ---------------------------------------------------------------------------
__global__ void gemm128_wmma(const float* __restrict__ X,
                             const float* __restrict__ W,
                             const float* __restrict__ bias,
                             float* __restrict__ Y, int M)
{
    __shared__ float wlds[DD * DD];           // 64 KB
    const int tid = threadIdx.x;

    // Cooperative W -> LDS copy: 4096 float4 by 256 threads (16 each).
    {
        const float4* __restrict__ wg = (const float4*)W;
        float4* wl = (float4*)wlds;
#pragma unroll
        for (int i = 0; i < 16; ++i)
            wl[tid + i * 256] = wg[tid + i * 256];
    }
    __syncthreads();

    const int lane = tid & 31;
    const int wave = tid >> 5;                // 0..7 -> row stripe
    const int lm   = lane & 15;               // 0..15
    const int lh   = lane >> 4;               // 0 or 1
    const int row0 = blockIdx.x * 128 + wave * 16;

    int arow = row0 + lm;
    if (arow >= M) arow = M - 1;              // clamp: duplicate read, store masked
    const float* aptr = X + (size_t)arow * DD + 2 * lh;

    // A fragments for this stripe: reused across all 8 column tiles.
    v2f afrag[32];
#pragma unroll
    for (int kk = 0; kk < 32; ++kk) {
        afrag[kk].x = aptr[4 * kk + 0];
        afrag[kk].y = aptr[4 * kk + 1];
    }

    for (int ct = 0; ct < 8; ++ct) {
        const int col0 = ct * 16;
        const float* bp = wlds + (2 * lh) * DD + col0 + lm;

        v8f c = {0.f, 0.f, 0.f, 0.f, 0.f, 0.f, 0.f, 0.f};
#pragma unroll
        for (int kk = 0; kk < 32; ++kk) {
            v2f b;
            b.x = bp[(4 * kk + 0) * DD];      // ds_load, conflict-free
            b.y = bp[(4 * kk + 1) * DD];
            c = __builtin_amdgcn_wmma_f32_16x16x4_f32(
                    /*neg_a=*/false, afrag[kk], /*neg_b=*/false, b,
                    /*c_mod=*/(short)0, c, /*reuse_a=*/false, /*reuse_b=*/false);
        }

        const float bc = bias ? bias[col0 + lm] : 0.f;
#pragma unroll
        for (int r = 0; r < 8; ++r) {
            const int row = row0 + r + 8 * lh;
            if (row < M) Y[(size_t)row * DD + col0 + lm] = c[r] + bc;
        }
    }
}

// ---------------------------------------------------------------------------
// Segment scatter-add: dstSum[dstIdx[e]][:] += src[srcIdx[e]][:]
//                      dstDeg[dstIdx[e]]   += 1
// 128 threads per incidence entry (one float per thread); L2 atomics.
// ---------------------------------------------------------------------------
__global__ void scatter_add(const float* __restrict__ src,
                            const int* __restrict__ dstIdx,
                            const int* __restrict__ srcIdx,
                            float* __restrict__ dstSum,
                            float* __restrict__ dstDeg, int nnz)
{
    const long long t = (long long)blockIdx.x * blockDim.x + threadIdx.x;
    const int e = (int)(t >> 7);
    const int d = (int)(t & 127);
    if (e >= nnz) return;
    const int s  = srcIdx[e];
    const int dd = dstIdx[e];
    atomicAdd(&dstSum[(size_t)dd * DD + d], src[(size_t)s * DD + d]);
    if (d == 0) atomicAdd(&dstDeg[dd], 1.0f);
}

__global__ void zero_f32(float* __restrict__ p, int n)
{
    const int t = blockIdx.x * blockDim.x + threadIdx.x;
    if (t < n) p[t] = 0.f;
}

// sum[t] /= max(deg[row],1)
__global__ void div_deg(float* __restrict__ sum, const float* __restrict__ deg, int n)
{
    const int t = blockIdx.x * blockDim.x + threadIdx.x;
    if (t >= n) return;
    sum[t] = sum[t] / fmaxf(deg[t >> 7], 1.f);
}

// sum[t] = sum[t]/max(deg[row],1) + bias[col]
__global__ void finalize_conv(float* __restrict__ sum, const float* __restrict__ deg,
                              const float* __restrict__ bias, int n)
{
    const int t = blockIdx.x * blockDim.x + threadIdx.x;
    if (t >= n) return;
    sum[t] = sum[t] / fmaxf(deg[t >> 7], 1.f) + bias[t & 127];
}

// ---------------------------------------------------------------------------
// Attention: one block per (head, query); online softmax over all keys.
// q,k,v are [rows,128] with head h occupying columns [16h,16h+16).
// Prefetch next key/value rows (global_prefetch_b8) to hide L2 latency.
// ---------------------------------------------------------------------------
__global__ void attn_kernel(const float* __restrict__ q,
                            const float* __restrict__ k,
                            const float* __restrict__ v,
                            float* __restrict__ out, int nkeys)
{
    const int h  = blockIdx.x & (HH - 1);
    const int qi = blockIdx.x >> 3;
    const int tid = threadIdx.x;

    float qv[16];
    const float* qp = q + (size_t)qi * DD + h * 16;
#pragma unroll
    for (int d = 0; d < 16; ++d) qv[d] = qp[d];

    float m = -INFINITY, l = 0.f;
    float acc[16];
#pragma unroll
    for (int d = 0; d < 16; ++d) acc[d] = 0.f;

    const size_t step = (size_t)blockDim.x * DD;
    for (int ki = tid; ki < nkeys; ki += blockDim.x) {
        const float* kp = k + (size_t)ki * DD + h * 16;
        const float* vp = v + (size_t)ki * DD + h * 16;
        __builtin_prefetch(kp + step, 0, 1);
        __builtin_prefetch(vp + step, 0, 1);
        float s = 0.f;
#pragma unroll
        for (int d = 0; d < 16; ++d) s += qv[d] * kp[d];
        s *= 0.25f;                                   // 1/sqrt(dh), dh=16
        const float mn = fmaxf(m, s);
        const float sc = __expf(m - mn);              // exp(-inf)=0 on first hit
        const float p  = __expf(s - mn);
        l = l * sc + p;
#pragma unroll
        for (int d = 0; d < 16; ++d) acc[d] = acc[d] * sc + p * vp[d];
        m = mn;
    }

    __shared__ float sm[256];
    __shared__ float sl[256];
    __shared__ float sa[256 * 16];
    sm[tid] = m; sl[tid] = l;
#pragma unroll
    for (int d = 0; d < 16; ++d) sa[tid * 16 + d] = acc[d];
    __syncthreads();

    for (int off = 128; off > 0; off >>= 1) {
        if (tid < off) {
            const float m2 = sm[tid + off], l2 = sl[tid + off];
            const float mn = fmaxf(sm[tid], m2);
            const float s1 = __expf(sm[tid] - mn);
            const float s2 = __expf(m2 - mn);
            sl[tid] = sl[tid] * s1 + l2 * s2;
#pragma unroll
            for (int d = 0; d < 16; ++d)
                sa[tid * 16 + d] = sa[tid * 16 + d] * s1 + sa[(tid + off) * 16 + d] * s2;
            sm[tid] = mn;
        }
        __syncthreads();
    }
    if (tid < 16) out[(size_t)qi * DD + h * 16 + tid] = sa[tid] / sl[0];
}

// ---------------------------------------------------------------------------
// SelfAttentionBatch pieces.
// rowscore: e[i] = tanh(h[i] @ A) . b   (grid = rows, block = 128)
// ---------------------------------------------------------------------------
__global__ void rowscore(const float* __restrict__ hmat, const float* __restrict__ A,
                         const float* __restrict__ bvec, float* __restrict__ e)
{
    const int i = blockIdx.x;
    const int j = threadIdx.x;
    float s = 0.f;
    for (int d = 0; d < DD; ++d) s += hmat[(size_t)i * DD + d] * A[(size_t)d * DD + j];
    __shared__ float red[DD];
    red[j] = tanhf(s) * bvec[j];
    __syncthreads();
    for (int off = 64; off > 0; off >>= 1) {
        if (j < off) red[j] += red[j + off];
        __syncthreads();
    }
    if (j == 0) e[i] = red[0];
}

// out[d] = sum_i softmax(e)_i * h[i,d]   (1 block, 128 threads)
__global__ void softmax_wsum(const float* __restrict__ e, const float* __restrict__ hmat,
                             float* __restrict__ out, int rows)
{
    const int d = threadIdx.x;
    float mx = -INFINITY;
    for (int i = 0; i < rows; ++i) mx = fmaxf(mx, e[i]);
    float sum = 0.f;
    for (int i = 0; i < rows; ++i) sum += __expf(e[i] - mx);
    float acc = 0.f;
    for (int i = 0; i < rows; ++i)
        acc += (__expf(e[i] - mx) / sum) * hmat[(size_t)i * DD + d];
    out[d] = acc;
}

// user = concat(context[50,128], u[1,128])
__global__ void assemble_user(const float* __restrict__ ctx, const float* __restrict__ u,
                              float* __restrict__ user)
{
    const int t = blockIdx.x * blockDim.x + threadIdx.x;
    if (t < CC * DD)            user[t] = ctx[t];
    else if (t < (CC + 1) * DD) user[t] = u[t - CC * DD];
}

// scores[j] = user_repr . rec_W[:,j] + rec_b[j]
__global__ void rec_scores(const float* __restrict__ ur, const float* __restrict__ W,
                           const float* __restrict__ b, float* __restrict__ out, int nent)
{
    __shared__ float u[DD];
    if (threadIdx.x < DD) u[threadIdx.x] = ur[threadIdx.x];
    __syncthreads();
    const int j = blockIdx.x * blockDim.x + threadIdx.x;
    if (j >= nent) return;
    float s = b[j];
    for (int d = 0; d < DD; ++d) s += u[d] * W[(size_t)d * nent + j];
    out[j] = s;
}

// ---------------------------------------------------------------------------
// Host-side orchestration
// ---------------------------------------------------------------------------
extern "C" void kernel_launch(void* const* d_in, const int* in_sizes, int n_in,
                              void* d_out, int out_size, void* d_ws, size_t ws_size,
                              hipStream_t stream)
{
    (void)in_sizes; (void)n_in; (void)out_size; (void)ws_size;

    const float* emb[3]   = {(const float*)d_in[0], (const float*)d_in[1], (const float*)d_in[2]};
    const float* ctx      =  (const float*)d_in[3];
    const float* theta[3] = {(const float*)d_in[4], (const float*)d_in[6], (const float*)d_in[8]};
    const float* bias[3]  = {(const float*)d_in[5], (const float*)d_in[7], (const float*)d_in[9]};
    const float* Wq = (const float*)d_in[10]; const float* bq = (const float*)d_in[11];
    const float* Wk = (const float*)d_in[12]; const float* bk = (const float*)d_in[13];
    const float* Wv = (const float*)d_in[14]; const float* bv = (const float*)d_in[15];
    const float* Wo = (const float*)d_in[16]; const float* bo = (const float*)d_in[17];
    const float* a_his = (const float*)d_in[18]; const float* b_his = (const float*)d_in[19];
    const float* a_kg  = (const float*)d_in[20]; const float* b_kg  = (const float*)d_in[21];
    const float* rec_W = (const float*)d_in[22]; const float* rec_b = (const float*)d_in[23];
    const int* nodes[3] = {(const int*)d_in[24], (const int*)d_in[26], (const int*)d_in[28]};
    const int* edges[3] = {(const int*)d_in[25], (const int*)d_in[27], (const int*)d_in[29]};

    // Workspace carve-up (floats). Total ~21.2M floats (~85 MB).
    float* ws = (float*)d_ws;
    size_t off = 0;
    float* xt    = ws + off; off += (size_t)NN * DD;       // X @ Theta
    float* esum  = ws + off; off += (size_t)EE * DD;       // hyperedge sums -> means
    float* edeg  = ws + off; off += EE;
    float* nsum  = ws + off; off += (size_t)NN * DD;       // node sums -> conv output
    float* ndeg  = ws + off; off += NN;
    float* kbuf  = ws + off; off += (size_t)3 * NN * DD;   // K projections of `related`
    float* vbuf  = ws + off; off += (size_t)3 * NN * DD;   // V projections
    float* qbuf  = ws + off; off += 64 * DD;
    float* aout  = ws + off; off += 64 * DD;               // attention output (pre-Wo)
    float* attv  = ws + off; off += 64 * DD;               // attentive = aout @ Wo + bo
    float* e_his = ws + off; off += 64;
    float* u_vec = ws + off; off += DD;
    float* user  = ws + off; off += (CC + 1) * DD;
    float* e_kg  = ws + off; off += 64;
    float* urepr = ws + off; off += DD;

    const int zcnt = EE * DD + EE + NN * DD + NN;          // esum..ndeg are contiguous
    const int gblk = (NN + 127) / 128;                     // 157 row blocks (128 rows each)

    for (int mI = 0; mI < 3; ++mI) {
        zero_f32<<<(zcnt + 255) / 256, 256, 0, stream>>>(esum, zcnt);
        // xt = X @ Theta
        gemm128_wmma<<<gblk, 256, 0, stream>>>(emb[mI], theta[mI], nullptr, xt, NN);
        // e_sum[edges] += xt[nodes];  e_deg[edges] += 1
        scatter_add<<<(int)(((long long)NNZ_ * DD + 255) / 256), 256, 0, stream>>>(
            xt, edges[mI], nodes[mI], esum, edeg, NNZ_);
        // m = e_sum / max(e_deg,1)  (in place)
        div_deg<<<(EE * DD + 255) / 256, 256, 0, stream>>>(esum, edeg, EE * DD);
        // n_sum[nodes] += m[edges];  n_deg[nodes] += 1
        scatter_add<<<(int)(((long long)NNZ_ * DD + 255) / 256), 256, 0, stream>>>(
            esum, nodes[mI], edges[mI], nsum, ndeg, NNZ_);
        // hout = n_sum / max(n_deg,1) + bias  (in place in nsum)
        finalize_conv<<<(NN * DD + 255) / 256, 256, 0, stream>>>(nsum, ndeg, bias[mI], NN * DD);
        // K/V projections of this modality's chunk of `related`
        gemm128_wmma<<<gblk, 256, 0, stream>>>(nsum, Wk, bk, kbuf + (size_t)mI * NN * DD, NN);
        gemm128_wmma<<<gblk, 256, 0, stream>>>(nsum, Wv, bv, vbuf + (size_t)mI * NN * DD, NN);
    }

    // Q projection of context entities
    gemm128_wmma<<<1, 256, 0, stream>>>(ctx, Wq, bq, qbuf, CC);
    // Multi-head attention over 3N keys (online softmax)
    attn_kernel<<<HH * CC, 256, 0, stream>>>(qbuf, kbuf, vbuf, aout, 3 * NN);
    // Output projection
    gemm128_wmma<<<1, 256, 0, stream>>>(aout, Wo, bo, attv, CC);

    // self_attn(attentive, a_his, b_his) -> u
    rowscore<<<CC, DD, 0, stream>>>(attv, a_his, b_his, e_his);
    softmax_wsum<<<1, DD, 0, stream>>>(e_his, attv, u_vec, CC);

    // user = concat(context, u); self_attn(user, a_kg, b_kg) -> user_repr
    assemble_user<<<((CC + 1) * DD + 255) / 256, 256, 0, stream>>>(ctx, u_vec, user);
    rowscore<<<CC + 1, DD, 0, stream>>>(user, a_kg, b_kg, e_kg);
    softmax_wsum<<<1, DD, 0, stream>>>(e_kg, user, urepr, CC + 1);

    // scores = user_repr @ rec_W + rec_b
    rec_scores<<<(NENT_ + 255) / 256, 256, 0, stream>>>(urepr, rec_W, rec_b,
                                                        (float*)d_out, NENT_);
}